// SynchronizationModule_15685220565449
// MI455X (gfx1250) — compile-verified
//
#include <hip/hip_runtime.h>

#define EPS_F 1e-8f

namespace {
constexpr int kB        = 16;
constexpr int kT        = 512;
constexpr int kD        = 2048;
constexpr int kN        = 4096;
constexpr int kSegs     = 8;
constexpr int kSegLen   = kT / kSegs;              // 64 rows per segment
constexpr int kRowsPerStage = 4;                   // rows DMA'd per pipeline stage
constexpr int kStages   = kSegLen / kRowsPerStage; // 16
constexpr int kThreads  = 1024;                    // 32 waves; one n per thread
constexpr int kNChunks  = kN / kThreads;           // 4
}

// Types matching the builtin signature (revealed by the round-1 diagnostic):
//   __builtin_amdgcn_global_load_async_to_lds_b64(v2i AS(1)*, v2i AS(3)*, Ii, Ii)
typedef int v2i __attribute__((vector_size(2 * sizeof(int))));
typedef __attribute__((address_space(1))) v2i g_v2i;   // global (prints as __device__)
typedef __attribute__((address_space(3))) v2i l_v2i;   // LDS

// Async DMA of 8 bytes/lane from global into LDS (gfx1250 async-load path).
__device__ __forceinline__ void async_copy8(const float* g, unsigned lds_off) {
#if __has_builtin(__builtin_amdgcn_global_load_async_to_lds_b64)
  __builtin_amdgcn_global_load_async_to_lds_b64(
      (g_v2i*)(__UINTPTR_TYPE__)g, (l_v2i*)(__UINTPTR_TYPE__)lds_off,
      /*offset=*/0, /*cpol=*/0);
#else
  asm volatile("global_load_async_to_lds_b64 %0, %1, off"
               :: "v"(lds_off), "v"((unsigned long long)(__UINTPTR_TYPE__)g)
               : "memory");
#endif
}

#if __has_builtin(__builtin_amdgcn_s_wait_asynccnt)
#define WAIT_ASYNC_LE(n) __builtin_amdgcn_s_wait_asynccnt(n)
#else
#define WAIT_ASYNC_LE(n) asm volatile("s_wait_asynccnt %0" :: "n"(n) : "memory")
#endif

// Defined FIRST so the disasm snippet shows the hot kernel (async DMA issue path).
__global__ __launch_bounds__(kThreads)
void decay_corr_main(const float* __restrict__ z_hist,
                     const int*   __restrict__ idx_i,
                     const int*   __restrict__ idx_j,
                     const float* __restrict__ ws,
                     float*       __restrict__ out) {
  __shared__ float rows[2][kRowsPerStage * kD];  // 2 x 32 KB double buffer

  const int tid = threadIdx.x;
  const int n   = blockIdx.x * kThreads + tid;   // one sampled pair per thread
  const int b   = blockIdx.y;
  const int seg = blockIdx.z;
  const int t0  = seg * kSegLen;
  const int t1  = t0 + kSegLen;

  const int   ii      = idx_i[n];
  const int   jj      = idx_j[n];
  const float r       = ws[n];
  const float d       = ws[kN + n];
  const float inv_den = ws[2 * kN + n];

  // Each thread DMAs 8 bytes (2 floats) of every row; 32 waves cover 8 KB/row.
  const float* gseg = z_hist + ((size_t)b * kT + t0) * kD + (size_t)tid * 2;
  const unsigned lbase0 = (unsigned)(__UINTPTR_TYPE__)&rows[0][tid * 2];
  const unsigned lbase1 = (unsigned)(__UINTPTR_TYPE__)&rows[1][tid * 2];

  // Prologue: stage 0 into buffer 0.
#pragma unroll
  for (int k = 0; k < kRowsPerStage; ++k)
    async_copy8(gseg + (size_t)k * kD, lbase0 + (unsigned)(k * kD * sizeof(float)));

  float acc = 0.0f;
  for (int s = 0; s < kStages; ++s) {
    const int cur = s & 1;
    if (s + 1 < kStages) {
      const float*   gnext = gseg + (size_t)(s + 1) * kRowsPerStage * kD;
      const unsigned lnext = cur ? lbase0 : lbase1;
#pragma unroll
      for (int k = 0; k < kRowsPerStage; ++k)
        async_copy8(gnext + (size_t)k * kD, lnext + (unsigned)(k * kD * sizeof(float)));
      WAIT_ASYNC_LE(kRowsPerStage);   // current stage's copies are done (in-order)
    } else {
      WAIT_ASYNC_LE(0);
    }
    __syncthreads();                  // all waves' slices of the stage visible

    const float* buf = rows[cur];
#pragma unroll
    for (int k = 0; k < kRowsPerStage; ++k) {
      // Horner recurrence: acc_t = acc_{t-1}*d + z_i*z_j  ==> Σ p_t d^{t1-1-t}
      const float zi = buf[k * kD + ii];
      const float zj = buf[k * kD + jj];
      acc = fmaf(acc, d, zi * zj);
    }
    __syncthreads();                  // safe to overwrite this buffer next stage
  }

  // Rescale segment partial from exponent (t1-1-t) to (T-1-t): multiply d^(T-t1).
  const float tail = expf(-r * (float)(kT - t1));
  atomicAdd(&out[(size_t)b * kN + n], acc * tail * inv_den);
}

// Per-n precompute: r = softplus(x), d = exp(-r), inv_den = 1/sqrt(sum_t d^k + eps)
__global__ void precompute_kernel(const float* __restrict__ decay_rates,
                                  float* __restrict__ ws) {
  int n = blockIdx.x * blockDim.x + threadIdx.x;
  if (n >= kN) return;
  float x = decay_rates[n];
  float r = fmaxf(x, 0.0f) + log1pf(expf(-fabsf(x)));  // stable softplus
  float d = expf(-r);
  float s = 0.0f;
#pragma unroll 8
  for (int t = 0; t < kT; ++t) s = fmaf(s, d, 1.0f);   // sum_{k=0}^{T-1} d^k
  ws[n]          = r;
  ws[kN + n]     = d;
  ws[2 * kN + n] = 1.0f / sqrtf(s + EPS_F);
}

__global__ void zero_out_kernel(float* __restrict__ out, int nel) {
  int i = blockIdx.x * blockDim.x + threadIdx.x;
  if (i < nel) out[i] = 0.0f;
}

extern "C" void kernel_launch(void* const* d_in, const int* in_sizes, int n_in,
                              void* d_out, int out_size, void* d_ws, size_t ws_size,
                              hipStream_t stream) {
  const float* z_hist      = (const float*)d_in[0];
  const float* decay_rates = (const float*)d_in[1];
  const int*   idx_i       = (const int*)d_in[2];
  const int*   idx_j       = (const int*)d_in[3];
  float* out = (float*)d_out;
  float* ws  = (float*)d_ws;   // uses 3*kN*4 = 48 KB

  zero_out_kernel<<<(out_size + 255) / 256, 256, 0, stream>>>(out, out_size);
  precompute_kernel<<<(kN + 255) / 256, 256, 0, stream>>>(decay_rates, ws);

  dim3 grid(kNChunks, kB, kSegs);   // 4 x 16 x 8 = 512 workgroups
  decay_corr_main<<<grid, kThreads, 0, stream>>>(z_hist, idx_i, idx_j, ws, out);
}